// GraphAttentionLayer_240518169172
// MI455X (gfx1250) — compile-verified
//
#include <hip/hip_runtime.h>
#include <hip/hip_bf16.h>
#include <float.h>

typedef __attribute__((ext_vector_type(2))) float v2f;
typedef __attribute__((ext_vector_type(8))) float v8f;

#define BB   2
#define CCH  64
#define HH   80
#define WWID 80
#define NN   (HH*WWID)      // 6400
#define NTILES (NN/16)      // 400
#define BN_EPS 1e-5f

// workspace layout (float offsets)
#define WS_Y     0                   // B*16*N raw conv outputs (rows 0-7 alpha, 8-15 sigma)
#define WS_SUM   (BB*16*NN)          // 16
#define WS_SQ    (WS_SUM+16)         // 16
#define WS_SCALE (WS_SQ+16)          // 16
#define WS_SHIFT (WS_SCALE+16)       // 16
#define WS_CDOT  (WS_SHIFT+16)       // B*(C-1)=126

// ---------------- kernel 0: zero BN accumulators ----------------
__global__ void k_init(float* __restrict__ ws) {
    if (threadIdx.x < 32) ws[WS_SUM + threadIdx.x] = 0.f;
}

// ---------------- kernel 1: fused dual 1x1 conv via WMMA f32 16x16x4 ----------------
// A = [16 x 64]: rows 0-7 = w_alpha, rows 8-15 = w_sigma.  B = x tile [64 x 16 nodes].
// One wave32 per 16-node output tile.  Also accumulates BN sum/sumsq per channel.
__global__ __launch_bounds__(32) void k_conv(const float* __restrict__ x,
                                             const float* __restrict__ wA,
                                             const float* __restrict__ wS,
                                             float* __restrict__ ws) {
    const int lane = threadIdx.x;
    const int b    = blockIdx.x / NTILES;
    const int tile = blockIdx.x % NTILES;
    const int row  = lane & 15;          // A: M index / B: N index
    const int koff = (lane >> 4) * 2;    // K sub-offset for this half-wave
    const int node = tile * 16 + row;

    const float* __restrict__ ap = (row < 8) ? (wA + row * CCH) : (wS + (row - 8) * CCH);
    const float* __restrict__ xb = x + (size_t)b * CCH * NN;

    v8f acc = {};
    #pragma unroll
    for (int k0 = 0; k0 < CCH; k0 += 4) {
        v2f a, bm;
        a.x  = ap[k0 + koff];
        a.y  = ap[k0 + koff + 1];
        bm.x = xb[(size_t)(k0 + koff)     * NN + node];
        bm.y = xb[(size_t)(k0 + koff + 1) * NN + node];
        acc = __builtin_amdgcn_wmma_f32_16x16x4_f32(
                  false, a, false, bm, (short)0, acc, false, false);
    }

    // D layout: lanes 0-15 hold rows 0-7 (alpha), lanes 16-31 rows 8-15 (sigma)
    float* __restrict__ yws = ws + WS_Y + (size_t)b * 16 * NN;
    const int chbase = (lane >> 4) * 8;
    #pragma unroll
    for (int v = 0; v < 8; ++v) {
        float val = acc[v];
        yws[(size_t)(chbase + v) * NN + node] = val;
        float s = val, q = val * val;
        // reduce across the 16 lanes of this half-wave
        #pragma unroll
        for (int m = 1; m < 16; m <<= 1) {
            s += __shfl_xor(s, m, 32);
            q += __shfl_xor(q, m, 32);
        }
        if (row == 0) {
            atomicAdd(ws + WS_SUM + chbase + v, s);
            atomicAdd(ws + WS_SQ  + chbase + v, q);
        }
    }
}

// ---------------- kernel 2: finalize BN affine params ----------------
__global__ void k_stats(const float* __restrict__ gA, const float* __restrict__ bA,
                        const float* __restrict__ gS, const float* __restrict__ bS,
                        float* __restrict__ ws) {
    int t = threadIdx.x;
    if (t >= 16) return;
    const float cnt = (float)(BB * NN);
    float mean = ws[WS_SUM + t] / cnt;
    float var  = ws[WS_SQ + t] / cnt - mean * mean;   // biased, torch BN style
    float inv  = rsqrtf(var + BN_EPS);
    float gamma = (t < 8) ? gA[t] : gS[t - 8];
    float beta  = (t < 8) ? bA[t] : bS[t - 8];
    float sc = gamma * inv;
    ws[WS_SCALE + t] = sc;
    ws[WS_SHIFT + t] = beta - mean * sc;
}

// ---------------- kernel 3: channel-pair dot products Gc[c,c+1] ----------------
__global__ __launch_bounds__(256) void k_cdot(const float* __restrict__ x,
                                              float* __restrict__ ws) {
    int b = blockIdx.x / (CCH - 1);
    int c = blockIdx.x % (CCH - 1);
    const float* __restrict__ p0 = x + ((size_t)b * CCH + c) * NN;
    const float* __restrict__ p1 = p0 + NN;
    float s = 0.f;
    for (int n = threadIdx.x; n < NN; n += 256) s += p0[n] * p1[n];
    __shared__ float red[256];
    red[threadIdx.x] = s;
    __syncthreads();
    for (int m = 128; m > 0; m >>= 1) {
        if (threadIdx.x < m) red[threadIdx.x] += red[threadIdx.x + m];
        __syncthreads();
    }
    if (threadIdx.x == 0) ws[WS_CDOT + blockIdx.x] = red[0];
}

// ---------------- kernel 4: fused spatial stencil softmax + channel blend ----------------
// one block (64 threads = 64 channels) per (batch, node)
__global__ __launch_bounds__(64) void k_final(const float* __restrict__ x,
                                              const float* __restrict__ ws,
                                              float* __restrict__ out) {
    const int b = blockIdx.x / NN;
    const int n = blockIdx.x % NN;
    const int r = n / WWID, c = n % WWID;
    const int t = threadIdx.x;

    __shared__ float lg[8];
    __shared__ int   nb[8];

    const float* __restrict__ yws = ws + WS_Y + (size_t)b * 16 * NN;
    const float* __restrict__ scl = ws + WS_SCALE;
    const float* __restrict__ shf = ws + WS_SHIFT;

    if (t < 8) {
        const int drs[8] = {-1,-1,-1, 0, 0, 1, 1, 1};
        const int dcs[8] = {-1, 0, 1,-1, 1,-1, 0, 1};
        int rr = r + drs[t], c2 = c + dcs[t];
        if (rr >= 0 && rr < HH && c2 >= 0 && c2 < WWID) {
            int m = rr * WWID + c2;
            float s = 0.f;
            #pragma unroll
            for (int k = 0; k < 8; ++k) {
                float a = fmaxf(yws[(size_t)k       * NN + n] * scl[k]     + shf[k],     0.f);
                float g = fmaxf(yws[(size_t)(8 + k) * NN + m] * scl[8 + k] + shf[8 + k], 0.f);
                s += a * g;
            }
            lg[t] = s; nb[t] = m;
        } else {
            lg[t] = -FLT_MAX; nb[t] = -1;
        }
    }
    __syncthreads();

    // masked softmax over <=8 neighbors (invalid -> exact 0, same as -9e15 masking)
    float mx = -FLT_MAX;
    #pragma unroll
    for (int j = 0; j < 8; ++j) mx = fmaxf(mx, lg[j]);
    float e[8], wsum = 0.f;
    #pragma unroll
    for (int j = 0; j < 8; ++j) {
        e[j] = (nb[j] >= 0) ? expf(lg[j] - mx) : 0.f;
        wsum += e[j];
    }
    const float inv = 1.f / wsum;

    const float* __restrict__ xb = x + (size_t)b * CCH * NN;
    float hs = 0.f;
    #pragma unroll
    for (int j = 0; j < 8; ++j) {
        int m = nb[j];
        if (m >= 0) hs += e[j] * xb[(size_t)t * NN + m];
    }
    hs *= inv;

    // channel GAT: chain adjacency -> softmax over {Gc[t,t-1], Gc[t,t+1]}
    const float* __restrict__ cd = ws + WS_CDOT + b * (CCH - 1);
    float gp = (t > 0)       ? cd[t - 1] : -FLT_MAX;
    float gn = (t < CCH - 1) ? cd[t]     : -FLT_MAX;
    float m2 = fmaxf(gp, gn);
    float ep = (t > 0)       ? expf(gp - m2) : 0.f;
    float en = (t < CCH - 1) ? expf(gn - m2) : 0.f;
    float i2 = 1.f / (ep + en);
    float hp = 0.f;
    if (t > 0)       hp += ep * xb[(size_t)(t - 1) * NN + n];
    if (t < CCH - 1) hp += en * xb[(size_t)(t + 1) * NN + n];
    hp *= i2;

    out[((size_t)b * CCH + t) * NN + n] = hs + hp;
}

extern "C" void kernel_launch(void* const* d_in, const int* in_sizes, int n_in,
                              void* d_out, int out_size, void* d_ws, size_t ws_size,
                              hipStream_t stream) {
    const float* x  = (const float*)d_in[0];
    const float* wA = (const float*)d_in[1];
    const float* gA = (const float*)d_in[2];
    const float* bA = (const float*)d_in[3];
    const float* wS = (const float*)d_in[4];
    const float* gS = (const float*)d_in[5];
    const float* bS = (const float*)d_in[6];
    float* out = (float*)d_out;
    float* ws  = (float*)d_ws;

    k_init <<<1, 32, 0, stream>>>(ws);
    k_conv <<<BB * NTILES, 32, 0, stream>>>(x, wA, wS, ws);
    k_stats<<<1, 32, 0, stream>>>(gA, bA, gS, bS, ws);
    k_cdot <<<BB * (CCH - 1), 256, 0, stream>>>(x, ws);
    k_final<<<BB * NN, 64, 0, stream>>>(x, ws, out);
}